// Clique_GraphConv_13065290514691
// MI455X (gfx1250) — compile-verified
//
#include <hip/hip_runtime.h>
#include <hip/hip_bf16.h>

#define DIM 128
#define DIM2 256
#define LDS_STRIDE 264   // 256 + 8 floats pad; 264*4 = 1056 B = multiple of 16
#define ROWS 32          // rows per block = 2 row-tiles of 16 (100000/32 = 3125 exact)

typedef __attribute__((ext_vector_type(2))) float v2f;
typedef __attribute__((ext_vector_type(4))) float v4f;
typedef __attribute__((ext_vector_type(8))) float v8f;

// ---------------------------------------------------------------------------
// Kernel 1: zero the workspace accumulators (summed[N*128] ++ deg[N]),
// vectorized b128 stores. total is a multiple of 4 (N*128 + N, N=100000).
// ---------------------------------------------------------------------------
__global__ void zero_ws_kernel(v4f* __restrict__ ws, int total4) {
    int i = blockIdx.x * blockDim.x + threadIdx.x;
    if (i < total4) ws[i] = (v4f){0.0f, 0.0f, 0.0f, 0.0f};
}

// ---------------------------------------------------------------------------
// Kernel 2: edge scatter. One wave32 per edge: 32 lanes x float4 = 128 floats.
// X row gather is a fully coalesced 512B read (X = 51.2MB, L2-resident in the
// 192MB L2). Scatter resolves in L2's atomic units (global_atomic_add_f32).
// ---------------------------------------------------------------------------
__global__ void __launch_bounds__(256)
scatter_edges_kernel(const float* __restrict__ X,
                     const int* __restrict__ src,
                     const int* __restrict__ dst,
                     float* __restrict__ summed,
                     float* __restrict__ deg,
                     int E) {
    int gid  = blockIdx.x * blockDim.x + threadIdx.x;
    int edge = gid >> 5;
    int lane = threadIdx.x & 31;
    if (edge >= E) return;

    int s = src[edge];
    int d = dst[edge];

    v4f m = *(const v4f*)(X + (size_t)s * DIM + lane * 4);
    float* p = summed + (size_t)d * DIM + lane * 4;
    atomicAdd(p + 0, m.x);
    atomicAdd(p + 1, m.y);
    atomicAdd(p + 2, m.z);
    atomicAdd(p + 3, m.w);
    if (lane == 0) atomicAdd(deg + d, 1.0f);
}

// ---------------------------------------------------------------------------
// Kernel 3: fused normalize + concat + GEMM + bias + ReLU.
// Block = 256 threads (8 waves) computes out[row0 : row0+32, 0:128].
// X-half of the H tile staged via async global->LDS b128 copies (ASYNCcnt);
// neighbor half (summed/max(deg,1)) staged on the VALU path, overlapping.
// Each wave owns a 16-wide column slice and 2 row-tiles: each B load feeds
// two V_WMMA_F32_16X16X4_F32, halving W's L2 traffic vs 16-row blocks.
// ---------------------------------------------------------------------------
__global__ void __launch_bounds__(256)
gemm_relu_kernel(const float* __restrict__ X,
                 const float* __restrict__ summed,
                 const float* __restrict__ deg,
                 const float* __restrict__ W,
                 const float* __restrict__ bias,
                 float* __restrict__ out) {
    __shared__ float sH[ROWS * LDS_STRIDE];   // 33,792 B

    const int row0 = blockIdx.x * ROWS;
    const int tid  = threadIdx.x;

    // ---- stage X half: 32 rows x 32 b128 quads, async global -> LDS -------
    for (int task = tid; task < ROWS * 32; task += 256) {
        int r = task >> 5;         // row in tile
        int q = task & 31;         // 16B quad within 128-float row
        unsigned lds_off = (unsigned)(size_t)(&sH[r * LDS_STRIDE + q * 4]);
        const float* gp = X + (size_t)(row0 + r) * DIM + q * 4;
        asm volatile("global_load_async_to_lds_b128 %0, %1, off"
                     :: "v"(lds_off), "v"(gp)
                     : "memory");
    }

    // ---- stage neighbor half: summed/max(deg,1), VALU path (overlaps) -----
    for (int task = tid; task < ROWS * 32; task += 256) {
        int r = task >> 5;
        int q = task & 31;
        int g = row0 + r;
        float inv = 1.0f / fmaxf(deg[g], 1.0f);
        v4f ns = *(const v4f*)(summed + (size_t)g * DIM + q * 4);
        ns.x *= inv; ns.y *= inv; ns.z *= inv; ns.w *= inv;
        *(v4f*)&sH[r * LDS_STRIDE + DIM + q * 4] = ns;
    }

    asm volatile("s_wait_asynccnt 0x0" ::: "memory");
    __syncthreads();

    // ---- WMMA K-loop -------------------------------------------------------
    const int lane = tid & 31;
    const int half = lane >> 4;       // 0: K={0,1}, 1: K={2,3}
    const int m    = lane & 15;       // A row / B column within 16-tile
    const int n    = (tid >> 5) * 16 + m;  // global output column 0..127

    v8f acc0 = {};
    v8f acc1 = {};
    #pragma unroll 4
    for (int k0 = 0; k0 < DIM2; k0 += 4) {
        // B 4x16 f32: half-wave K split, column n  (shared by both row-tiles)
        v2f b;
        b.x = W[(size_t)(k0 + half * 2)     * DIM + n];
        b.y = W[(size_t)(k0 + half * 2 + 1) * DIM + n];
        // A 16x4 f32 for row-tile 0 and 1
        v2f a0 = *(const v2f*)&sH[(m     ) * LDS_STRIDE + k0 + half * 2];
        v2f a1 = *(const v2f*)&sH[(m + 16) * LDS_STRIDE + k0 + half * 2];
        acc0 = __builtin_amdgcn_wmma_f32_16x16x4_f32(
                   false, a0, false, b, (short)0, acc0, false, false);
        acc1 = __builtin_amdgcn_wmma_f32_16x16x4_f32(
                   false, a1, false, b, (short)0, acc1, false, false);
    }

    // ---- epilogue: bias + ReLU + store ------------------------------------
    // C/D layout: VGPR i -> row M = half*8 + i, column N = m (per 16-lane half)
    float bn = bias[n];
    float* op0 = out + (size_t)(row0      + half * 8) * DIM + n;
    float* op1 = out + (size_t)(row0 + 16 + half * 8) * DIM + n;
    #pragma unroll
    for (int i = 0; i < 8; i++) {
        op0[(size_t)i * DIM] = fmaxf(acc0[i] + bn, 0.0f);
        op1[(size_t)i * DIM] = fmaxf(acc1[i] + bn, 0.0f);
    }
}

// ---------------------------------------------------------------------------
extern "C" void kernel_launch(void* const* d_in, const int* in_sizes, int n_in,
                              void* d_out, int out_size, void* d_ws, size_t ws_size,
                              hipStream_t stream) {
    const float* X    = (const float*)d_in[0];
    const int*   src  = (const int*)d_in[1];
    const int*   dst  = (const int*)d_in[2];
    const float* W    = (const float*)d_in[3];
    const float* bias = (const float*)d_in[4];
    float*       out  = (float*)d_out;

    const int N = in_sizes[0] / DIM;   // 100000
    const int E = in_sizes[1];         // 3200000

    float* summed = (float*)d_ws;                 // N*128 floats
    float* deg    = summed + (size_t)N * DIM;     // N floats

    // 1) zero accumulators (workspace is poisoned each run)
    int total4 = (N * DIM + N) / 4;
    zero_ws_kernel<<<(total4 + 255) / 256, 256, 0, stream>>>((v4f*)d_ws, total4);

    // 2) edge scatter: 8 edges per 256-thread block (one wave per edge)
    scatter_edges_kernel<<<(E + 7) / 8, 256, 0, stream>>>(X, src, dst, summed, deg, E);

    // 3) fused normalize + concat + GEMM + bias + ReLU (WMMA f32, async staging)
    gemm_relu_kernel<<<N / ROWS, 256, 0, stream>>>(X, summed, deg, W, bias, out);
}